// MambaMoE_24378234372445
// MI455X (gfx1250) — compile-verified
//
#include <hip/hip_runtime.h>
#include <hip/hip_bf16.h>
#include <math.h>

#define B_       2
#define L_       1024
#define DIM_     1024
#define DEPTH_   2
#define D_STATE_ 64
#define D_CONV_  4
#define D_INNER_ 2048
#define DT_RANK_ 64
#define NE_      8
#define E_DIM_   2048
#define T_       (B_*L_)      // 2048 tokens
#define XPROJ_N_ (DT_RANK_ + 2*D_STATE_)  // 192
#define KC_      64           // K-chunk staged in LDS per TDM transfer

typedef __attribute__((ext_vector_type(2))) float        v2f;
typedef __attribute__((ext_vector_type(8))) float        v8f;
typedef __attribute__((ext_vector_type(4))) unsigned int v4u;
typedef __attribute__((ext_vector_type(8))) int          v8i;
typedef __attribute__((ext_vector_type(4))) int          v4i;

__device__ __forceinline__ float sigmoidf_(float x){ return 1.f/(1.f+__expf(-x)); }
__device__ __forceinline__ float siluf_(float x){ return x*sigmoidf_(x); }
__device__ __forceinline__ float softplusf_(float x){ return (x>20.f)?x:log1pf(__expf(x)); }
__device__ __forceinline__ float geluf_(float x){ return 0.5f*x*(1.f+erff(x*0.70710678118654752f)); }

// ---------------------------------------------------------------------------
// Tensor Data Mover: DMA a 2-D fp32 tile (tile_w x tile_h elements) from
// global (row stride = stride elements) into LDS at byte offset lds_off.
// D# bit packing per CDNA5 ISA 8.3/8.4:
//  g0: count=1 | lds_addr | global_addr[56:0] | type=2
//  g1: data_size=4B | tensor_dim0/1 (OOB zero-fill) | tile_dim0/1 | dim0 stride
// Rows outside tensor_h read back as zero (handles ragged N tiles).
// This toolchain's builtin takes 6 args: (g0, g1, g2, g3, g4, cpol).
// ---------------------------------------------------------------------------
__device__ __forceinline__ void tdm_load_2d(unsigned lds_off, const float* gptr,
                                            unsigned tile_w, unsigned tile_h,
                                            unsigned tens_w, unsigned tens_h,
                                            unsigned long long stride_elems)
{
  unsigned long long ga = (unsigned long long)(uintptr_t)gptr;
  v4u g0;
  g0.x = 1u;                                              // count=1, user mode
  g0.y = lds_off;                                         // lds_addr
  g0.z = (unsigned)ga;                                    // global_addr[31:0]
  g0.w = (unsigned)((ga >> 32) & 0x01FFFFFFu) | (2u << 30); // addr[56:32] | type=2
  v8i g1;
  g1[0] = (int)(2u << 16);                                // data_size = 4 bytes
  g1[1] = (int)((tens_w & 0xFFFFu) << 16);                // tensor_dim0[15:0]
  g1[2] = (int)((tens_w >> 16) | ((tens_h & 0xFFFFu) << 16));
  g1[3] = (int)((tens_h >> 16) | ((tile_w & 0xFFFFu) << 16));
  g1[4] = (int)(tile_h & 0xFFFFu);                        // tile_dim1, tile_dim2=0
  g1[5] = (int)(unsigned)(stride_elems & 0xFFFFFFFFull);  // dim0_stride[31:0]
  g1[6] = (int)(unsigned)((stride_elems >> 32) & 0xFFFFull);
  g1[7] = 0;
  v4i gz4 = {0,0,0,0};
  v8i gz8 = {0,0,0,0,0,0,0,0};
  __builtin_amdgcn_tensor_load_to_lds(g0, g1, gz4, gz4, gz8, 0);
}

// ---------------------------------------------------------------------------
// fp32 WMMA GEMM, TDM-fed:  C[M,N] = act(A[M,K] @ W[N,K]^T + bias) (+ addsrc)
// Block = 8 waves = 16(M) x 128(N) tile. K chunked by KC_; A (16xKC) and
// W (128xKC) tiles double-buffered in LDS via tensor_load_to_lds; wave 0
// issues DMA + s_wait_tensorcnt, barrier publishes. WMMA operands come from
// ds_load_b64. Per-wave 16x16 tile, V_WMMA_F32_16X16X4_F32.
// act: 0 = none, 1 = softplus
// ---------------------------------------------------------------------------
__global__ void __launch_bounds__(256) gemm_wmma_f32(
    const float* __restrict__ A, int lda,
    const float* __restrict__ W, int ldw,
    const float* __restrict__ bias,
    const float* __restrict__ addsrc,
    float* __restrict__ C, int ldc,
    int M, int N, int K, int act)
{
  __shared__ float lA[2][16*KC_];     // 2 x 4 KB
  __shared__ float lW[2][128*KC_];    // 2 x 32 KB
  const int lane = threadIdx.x & 31;
  const int wid  = threadIdx.x >> 5;
  const int tiles_n = N >> 4;
  const int tm   = blockIdx.y;                 // 16-row tile index
  const int nb0  = blockIdx.x * 128;           // first column of block
  const int tnb  = blockIdx.x*8 + wid;         // this wave's 16-col tile
  const bool valid = tnb < tiles_n;

  const unsigned ldsA = (unsigned)(uintptr_t)&lA[0][0];
  const unsigned ldsW = (unsigned)(uintptr_t)&lW[0][0];
  const unsigned wrem = (unsigned)((N - nb0) < 128 ? (N - nb0) : 128);
  const float* Abase = A + (size_t)tm*16*lda;
  const float* Wbase = W + (size_t)nb0*ldw;
  const int nc = K / KC_;

  if (wid == 0) {                               // prologue: fill buffer 0
    tdm_load_2d(ldsA, Abase, KC_, 16,  KC_, 16,   (unsigned long long)lda);
    tdm_load_2d(ldsW, Wbase, KC_, 128, KC_, wrem, (unsigned long long)ldw);
    __builtin_amdgcn_s_wait_tensorcnt(0);
  }
  __syncthreads();

  const int am = lane & 15;
  const int kb = (lane >> 4) << 1;
  v8f acc = {0.f,0.f,0.f,0.f,0.f,0.f,0.f,0.f};

  for (int c = 0; c < nc; ++c) {
    const int buf = c & 1;
    if (wid == 0 && (c+1) < nc) {               // DMA next chunk into other buf
      const unsigned boff = (unsigned)(buf ^ 1);
      tdm_load_2d(ldsA + boff*16*KC_*4,  Abase + (size_t)(c+1)*KC_,
                  KC_, 16,  KC_, 16,   (unsigned long long)lda);
      tdm_load_2d(ldsW + boff*128*KC_*4, Wbase + (size_t)(c+1)*KC_,
                  KC_, 128, KC_, wrem, (unsigned long long)ldw);
    }
    const float* ap = &lA[buf][am*KC_ + kb];
    const float* wp = &lW[buf][(wid*16 + am)*KC_ + kb];
    #pragma unroll
    for (int k = 0; k < KC_; k += 4) {
      v2f a = *(const v2f*)(ap + k);            // ds_load_b64
      v2f b = *(const v2f*)(wp + k);            // ds_load_b64
      acc = __builtin_amdgcn_wmma_f32_16x16x4_f32(false, a, false, b,
                                                  (short)0, acc, false, false);
    }
    if (wid == 0 && (c+1) < nc) __builtin_amdgcn_s_wait_tensorcnt(0);
    __syncthreads();                            // publish next buf / retire reads
  }

  if (valid) {
    const int half = lane >> 4;
    const int col  = tnb*16 + am;
    const float bv = bias ? bias[col] : 0.f;
    #pragma unroll
    for (int r = 0; r < 8; ++r) {
      const int row = tm*16 + r + 8*half;
      float v = acc[r] + bv;
      if (act == 1) v = softplusf_(v);
      if (addsrc) v += addsrc[(size_t)row*ldc + col];
      C[(size_t)row*ldc + col] = v;
    }
  }
}

// ---------------------------------------------------------------------------
// h = x + pos_emb  (pos_emb broadcast over batch)
// ---------------------------------------------------------------------------
__global__ void add_posemb(const float* __restrict__ x, const float* __restrict__ pe,
                           float* __restrict__ h)
{
  const size_t idx = (size_t)blockIdx.x*256 + threadIdx.x;   // over T_*DIM_
  const int d = (int)(idx & (DIM_-1));
  const int t = (int)(idx >> 10);
  const int l = t & (L_-1);
  h[idx] = x[idx] + pe[(size_t)l*DIM_ + d];
}

// ---------------------------------------------------------------------------
// causal depthwise conv1d (kernel 4) + SiLU.  u_raw = xz[:, 0:D_INNER]
// ---------------------------------------------------------------------------
__global__ void conv_silu(const float* __restrict__ xz, const float* __restrict__ cw,
                          const float* __restrict__ cb, float* __restrict__ u)
{
  const size_t idx = (size_t)blockIdx.x*256 + threadIdx.x;   // over T_*D_INNER_
  const int c = (int)(idx & (D_INNER_-1));
  const int t = (int)(idx >> 11);
  const int b = t >> 10, l = t & (L_-1);
  float acc = cb[c];
  #pragma unroll
  for (int k = 0; k < D_CONV_; ++k) {
    const int ls = l + k - (D_CONV_-1);
    if (ls >= 0)
      acc += xz[(size_t)(b*L_+ls)*(2*D_INNER_) + c] * cw[c*D_CONV_ + k];
  }
  u[idx] = siluf_(acc);
}

// ---------------------------------------------------------------------------
// Selective scan. One lane owns one (b,d) channel with all 64 states in VGPRs.
// B_t / C_t (shared across channels) staged through LDS per step.
// ---------------------------------------------------------------------------
__global__ void __launch_bounds__(256,1) scan_kernel(
    const float* __restrict__ u, const float* __restrict__ dt,
    const float* __restrict__ dbl, const float* __restrict__ A_log,
    float* __restrict__ ys)
{
  __shared__ float lB[D_STATE_];
  __shared__ float lC[D_STATE_];
  const int ch = blockIdx.x*256 + threadIdx.x;   // 0..4095
  const int b = ch >> 11;
  const int d = ch & (D_INNER_-1);
  float Aloc[D_STATE_], h[D_STATE_];
  #pragma unroll
  for (int s = 0; s < D_STATE_; ++s) {
    Aloc[s] = -__expf(A_log[(size_t)d*D_STATE_ + s]);
    h[s] = 0.f;
  }
  for (int t = 0; t < L_; ++t) {
    const int row = b*L_ + t;
    if (threadIdx.x < 64)
      lB[threadIdx.x] = dbl[(size_t)row*XPROJ_N_ + DT_RANK_ + threadIdx.x];
    else if (threadIdx.x < 128)
      lC[threadIdx.x-64] = dbl[(size_t)row*XPROJ_N_ + DT_RANK_ + D_STATE_ + (threadIdx.x-64)];
    __syncthreads();
    const float ut  = u [(size_t)row*D_INNER_ + d];
    const float dtt = dt[(size_t)row*D_INNER_ + d];
    const float dtu = dtt*ut;
    float y = 0.f;
    #pragma unroll
    for (int s = 0; s < D_STATE_; ++s) {
      h[s] = __expf(dtt*Aloc[s])*h[s] + dtu*lB[s];
      y += h[s]*lC[s];
    }
    ys[(size_t)row*D_INNER_ + d] = y;
    __syncthreads();
  }
}

// ---------------------------------------------------------------------------
// y = (ys + u*Dskip) * silu(z);  z = xz[:, D_INNER:2*D_INNER].  In-place on u.
// ---------------------------------------------------------------------------
__global__ void gate_kernel(const float* __restrict__ ys, const float* __restrict__ xz,
                            const float* __restrict__ Dskip, float* __restrict__ u)
{
  const size_t idx = (size_t)blockIdx.x*256 + threadIdx.x;   // over T_*D_INNER_
  const int c = (int)(idx & (D_INNER_-1));
  const size_t t = idx >> 11;
  const float z = xz[t*(2*D_INNER_) + D_INNER_ + c];
  u[idx] = (ys[idx] + u[idx]*Dskip[c]) * siluf_(z);
}

// ---------------------------------------------------------------------------
// Top-1 router: logits -> argmax -> compaction into per-expert token lists.
// ---------------------------------------------------------------------------
__global__ void route_kernel(const float* __restrict__ x, const float* __restrict__ rw,
                             const float* __restrict__ rb, int* __restrict__ sel,
                             int* __restrict__ cnt, int* __restrict__ toklist)
{
  __shared__ float lg[32][NE_];
  const int ti = threadIdx.x >> 3;
  const int e  = threadIdx.x & 7;
  const int t  = blockIdx.x*32 + ti;
  const float* xr = x  + (size_t)t*DIM_;
  const float* wr = rw + (size_t)e*DIM_;
  float acc = rb[e];
  for (int k = 0; k < DIM_; ++k) acc += xr[k]*wr[k];
  lg[ti][e] = acc;
  __syncthreads();
  if (e == 0) {
    float best = lg[ti][0]; int bi = 0;
    #pragma unroll
    for (int j = 1; j < NE_; ++j) { float v = lg[ti][j]; if (v > best) { best = v; bi = j; } }
    sel[t] = bi;
    const int pos = atomicAdd(&cnt[bi], 1);
    toklist[bi*T_ + pos] = t;
  }
}

__global__ void zero_ints(int* __restrict__ p, int n)
{
  const int i = blockIdx.x*256 + threadIdx.x;
  if (i < n) p[i] = 0;
}

// ---------------------------------------------------------------------------
// Grouped (gather) expert GEMM: one 16-token ragged tile of one expert x
// 128 columns per block. Pads for WMMA (EXEC all-1), predicates only stores.
// ---------------------------------------------------------------------------
__global__ void __launch_bounds__(256) moe_gemm(
    const float* __restrict__ X, int K,
    const float* __restrict__ Wt,       // [NE, N, K]
    const float* __restrict__ bias,     // [NE, N]
    const int*   __restrict__ cnt,
    const int*   __restrict__ toklist,  // [NE, T_]
    const float* __restrict__ res,      // [T_, N] or null
    float* __restrict__ Y, int N, int act)
{
  __shared__ int lidx[16];
  __shared__ int s_nt;
  const int e = blockIdx.z;
  if (threadIdx.x == 0) s_nt = cnt[e];
  __syncthreads();
  const int nt   = s_nt;
  const int base = blockIdx.y*16;
  if (base >= nt) return;                         // block-uniform
  if (threadIdx.x < 16) {
    const int g = base + threadIdx.x;
    lidx[threadIdx.x] = toklist[e*T_ + (g < nt ? g : nt-1)];
  }
  __syncthreads();
  const int lane = threadIdx.x & 31;
  const int wid  = threadIdx.x >> 5;
  const int nb   = blockIdx.x*128 + wid*16;
  const int kb   = (lane >> 4) << 1;
  const int myrow = lidx[lane & 15];
  const float* arow = X  + (size_t)myrow*K + kb;
  const float* wrow = Wt + ((size_t)e*N + (nb + (lane & 15)))*K + kb;
  v8f acc = {0.f,0.f,0.f,0.f,0.f,0.f,0.f,0.f};
  #pragma unroll 8
  for (int k = 0; k < K; k += 4) {
    __builtin_prefetch(wrow + k + 256, 0, 1);
    v2f a = *(const v2f*)(arow + k);
    v2f b = *(const v2f*)(wrow + k);
    acc = __builtin_amdgcn_wmma_f32_16x16x4_f32(false, a, false, b, (short)0, acc, false, false);
  }
  const int half = lane >> 4;
  const int col  = nb + (lane & 15);
  const float bv = bias[(size_t)e*N + col];
  #pragma unroll
  for (int r = 0; r < 8; ++r) {
    const int rit = r + 8*half;
    if (base + rit < nt) {                        // predicate stores only
      const int t = lidx[rit];
      float v = acc[r] + bv;
      if (act) v = geluf_(v);
      if (res) v += res[(size_t)t*N + col];
      Y[(size_t)t*N + col] = v;
    }
  }
}

__global__ void copy_out(const float* __restrict__ src, float* __restrict__ dst, int n)
{
  const int i = blockIdx.x*256 + threadIdx.x;
  if (i < n) dst[i] = src[i];
}

// ---------------------------------------------------------------------------
extern "C" void kernel_launch(void* const* d_in, const int* in_sizes, int n_in,
                              void* d_out, int out_size, void* d_ws, size_t ws_size,
                              hipStream_t stream)
{
  const float* x         = (const float*)d_in[0];
  const float* pos_emb   = (const float*)d_in[1];
  const float* in_proj_w = (const float*)d_in[2];
  const float* conv_w    = (const float*)d_in[3];
  const float* conv_b    = (const float*)d_in[4];
  const float* x_proj_w  = (const float*)d_in[5];
  const float* dt_proj_w = (const float*)d_in[6];
  const float* dt_proj_b = (const float*)d_in[7];
  const float* A_log     = (const float*)d_in[8];
  const float* Dskip     = (const float*)d_in[9];
  const float* out_proj_w= (const float*)d_in[10];
  const float* route_w   = (const float*)d_in[11];
  const float* route_b   = (const float*)d_in[12];
  const float* w1        = (const float*)d_in[13];
  const float* b1        = (const float*)d_in[14];
  const float* w2        = (const float*)d_in[15];
  const float* b2        = (const float*)d_in[16];

  // workspace carve-up (all fp32)
  float* ws  = (float*)d_ws;
  float* hA  = ws;  ws += (size_t)T_*DIM_;            // 2M
  float* hB  = ws;  ws += (size_t)T_*DIM_;            // 2M
  float* xz  = ws;  ws += (size_t)T_*2*D_INNER_;      // 8M  (reused as h1)
  float* u   = ws;  ws += (size_t)T_*D_INNER_;        // 4M  (gated in-place)
  float* dbl = ws;  ws += (size_t)T_*XPROJ_N_;        // 0.4M
  float* dtb = ws;  ws += (size_t)T_*D_INNER_;        // 4M
  float* ysb = ws;  ws += (size_t)T_*D_INNER_;        // 4M
  int*   sel = (int*)ws;        ws += T_;
  int*   cnt = (int*)ws;        ws += 32;
  int*   tok = (int*)ws;        ws += (size_t)NE_*T_;
  float* h1  = xz;                                    // alias: xz free after gate

  const int EW = (T_*D_INNER_) / 256;                 // elementwise grid

  add_posemb<<<(T_*DIM_)/256, 256, 0, stream>>>(x, pos_emb, hA);

  float* hcur = hA;   // layer input / moe output
  float* hoth = hB;   // mamba output (+residual) / moe residual source

  for (int i = 0; i < DEPTH_; ++i) {
    const float* ipw = in_proj_w + (size_t)i*(2*D_INNER_)*DIM_;
    const float* cw  = conv_w    + (size_t)i*D_INNER_*D_CONV_;
    const float* cb  = conv_b    + (size_t)i*D_INNER_;
    const float* xpw = x_proj_w  + (size_t)i*XPROJ_N_*D_INNER_;
    const float* dtw = dt_proj_w + (size_t)i*D_INNER_*DT_RANK_;
    const float* dtbp= dt_proj_b + (size_t)i*D_INNER_;
    const float* Al  = A_log     + (size_t)i*D_INNER_*D_STATE_;
    const float* Dk  = Dskip     + (size_t)i*D_INNER_;
    const float* opw = out_proj_w+ (size_t)i*DIM_*D_INNER_;
    const float* rw  = route_w   + (size_t)i*NE_*DIM_;
    const float* rb  = route_b   + (size_t)i*NE_;
    const float* w1p = w1 + (size_t)i*NE_*E_DIM_*DIM_;
    const float* b1p = b1 + (size_t)i*NE_*E_DIM_;
    const float* w2p = w2 + (size_t)i*NE_*DIM_*E_DIM_;
    const float* b2p = b2 + (size_t)i*NE_*DIM_;

    // xz = h @ in_proj_w^T                      [T, 4096]
    gemm_wmma_f32<<<dim3((2*D_INNER_)/128, T_/16), 256, 0, stream>>>(
        hcur, DIM_, ipw, DIM_, nullptr, nullptr, xz, 2*D_INNER_,
        T_, 2*D_INNER_, DIM_, 0);
    // u = silu(causal_conv(xz[:, :D_INNER]))
    conv_silu<<<EW, 256, 0, stream>>>(xz, cw, cb, u);
    // dbl = u @ x_proj_w^T                      [T, 192]
    gemm_wmma_f32<<<dim3((XPROJ_N_+127)/128, T_/16), 256, 0, stream>>>(
        u, D_INNER_, xpw, D_INNER_, nullptr, nullptr, dbl, XPROJ_N_,
        T_, XPROJ_N_, D_INNER_, 0);
    // dt = softplus(dbl[:, :64] @ dt_proj_w^T + dt_proj_b)   [T, 2048]
    gemm_wmma_f32<<<dim3(D_INNER_/128, T_/16), 256, 0, stream>>>(
        dbl, XPROJ_N_, dtw, DT_RANK_, dtbp, nullptr, dtb, D_INNER_,
        T_, D_INNER_, DT_RANK_, 1);
    // selective scan -> ys
    scan_kernel<<<(B_*D_INNER_)/256, 256, 0, stream>>>(u, dtb, dbl, Al, ysb);
    // u = (ys + u*Dskip) * silu(z)
    gate_kernel<<<EW, 256, 0, stream>>>(ysb, xz, Dk, u);
    // hoth = u @ out_proj_w^T + hcur   (mamba residual)
    gemm_wmma_f32<<<dim3(DIM_/128, T_/16), 256, 0, stream>>>(
        u, D_INNER_, opw, D_INNER_, nullptr, hcur, hoth, DIM_,
        T_, DIM_, D_INNER_, 0);

    // --- MoE (top-1: compute only the selected expert) ---
    zero_ints<<<1, 256, 0, stream>>>(cnt, NE_);
    route_kernel<<<T_/32, 256, 0, stream>>>(hoth, rw, rb, sel, cnt, tok);
    {
      dim3 g(E_DIM_/128, T_/16, NE_);
      moe_gemm<<<g, 256, 0, stream>>>(hoth, DIM_, w1p, b1p, cnt, tok,
                                      nullptr, h1, E_DIM_, 1);
    }
    {
      dim3 g(DIM_/128, T_/16, NE_);
      moe_gemm<<<g, 256, 0, stream>>>(h1, E_DIM_, w2p, b2p, cnt, tok,
                                      hoth, hcur, DIM_, 0);
    }
  }

  copy_out<<<(T_*DIM_+255)/256, 256, 0, stream>>>(hcur, (float*)d_out, T_*DIM_);
}